// UKF_74294344286896
// MI455X (gfx1250) — compile-verified
//
#include <hip/hip_runtime.h>
#include <hip/hip_bf16.h>

// ---------------------------------------------------------------------------
// UKF(N=2048) on MI455X. Symmetric sqrtm sigma points cancel -> linear Kalman
// step. All GEMMs are A @ B^T (every right operand is symmetric), computed
// with V_WMMA_F32_16X16X4_F32 through a double-buffered LDS pipeline fed by
// GLOBAL_LOAD_ASYNC_TO_LDS_B128 (ASYNCcnt / s_wait_asynccnt).
// ---------------------------------------------------------------------------

typedef __attribute__((ext_vector_type(2))) float v2f;
typedef __attribute__((ext_vector_type(8))) float v8f;

constexpr int NDIM  = 2048;
constexpr int BT    = 128;           // block tile (M and N)
constexpr int KC    = 16;            // K chunk staged through LDS
constexpr int LDSP  = 20;            // padded row stride (floats): 80B, 16B-aligned rows
constexpr int TILE  = BT * LDSP;     // one staged matrix tile (floats)
constexpr int STAGE = 2 * TILE;      // A tile + B tile (floats)
constexpr int NCH   = NDIM / KC;     // 128 K-chunks

// Out = alpha * A @ B^T + beta * Cin ; A,B,Cin,Out row-major [NDIM x NDIM].
__global__ __launch_bounds__(256) void gemm_wmma_nt(const float* __restrict__ A,
                                                    const float* __restrict__ B,
                                                    const float* __restrict__ Cin,
                                                    float* __restrict__ Out,
                                                    float alpha, float beta) {
  __shared__ float Sm[2 * STAGE];    // [buf][A(128x16) | B(128x16)], stride 20 -> 40KB

  const int tid  = threadIdx.x;
  const int lane = tid & 31;
  const int wave = tid >> 5;
  const int wm   = wave & 3;         // 4 waves along M -> 32 rows each
  const int wn   = wave >> 2;        // 2 waves along N -> 64 cols each
  const int l16  = lane & 15;
  const int lh   = lane >> 4;        // selects K pair within fragment

  const size_t bm = (size_t)blockIdx.y * BT;
  const size_t bn = (size_t)blockIdx.x * BT;

  // staging indices: 128 rows x 16 floats = 512 float4; 256 threads x 2
  const int srow = tid >> 2;         // 0..63  (+64 for second half)
  const int scol = (tid & 3) * 4;    // 0,4,8,12

  const unsigned ldsBase = (unsigned)(size_t)(void*)Sm;

  // issue async b128 loads of chunk starting at column kc2 into buffer `buf`
  auto stage = [&](int kc2, int buf) {
    const unsigned base = ldsBase + (unsigned)(buf * STAGE) * 4u;
#pragma unroll
    for (int i = 0; i < 2; ++i) {
      const int r = srow + 64 * i;
      const unsigned la = base + (unsigned)(r * LDSP + scol) * 4u;
      const float* ga = A + (bm + r) * NDIM + kc2 + scol;
      asm volatile("global_load_async_to_lds_b128 %0, %1, off"
                   :: "v"(la), "v"(ga) : "memory");
      const unsigned lb = base + (unsigned)(TILE + r * LDSP + scol) * 4u;
      const float* gb = B + (bn + r) * NDIM + kc2 + scol;
      asm volatile("global_load_async_to_lds_b128 %0, %1, off"
                   :: "v"(lb), "v"(gb) : "memory");
    }
  };

  v8f acc[2][4];
#pragma unroll
  for (int mt = 0; mt < 2; ++mt)
#pragma unroll
    for (int nt = 0; nt < 4; ++nt) acc[mt][nt] = (v8f)0.0f;

  stage(0, 0);  // prefetch first chunk

  for (int i = 0; i < NCH; ++i) {
    __syncthreads();  // everyone done reading the buffer we are about to fill
    if (i + 1 < NCH) {
      stage((i + 1) * KC, (i + 1) & 1);
      asm volatile("s_wait_asynccnt 0x4" ::: "memory");  // chunk i landed
    } else {
      asm volatile("s_wait_asynccnt 0x0" ::: "memory");
    }
    __syncthreads();  // chunk i visible to all waves

    const float* Ab = &Sm[(i & 1) * STAGE];
    const float* Bb = Ab + TILE;

#pragma unroll
    for (int kk = 0; kk < KC; kk += 4) {
      const int kp = kk + 2 * lh;  // even -> 8B aligned ds_load_b64
      const v2f a0 = *(const v2f*)&Ab[(wm * 32 + l16) * LDSP + kp];
      const v2f a1 = *(const v2f*)&Ab[(wm * 32 + 16 + l16) * LDSP + kp];
#pragma unroll
      for (int nt = 0; nt < 4; ++nt) {
        const v2f b = *(const v2f*)&Bb[(wn * 64 + nt * 16 + l16) * LDSP + kp];
        acc[0][nt] = __builtin_amdgcn_wmma_f32_16x16x4_f32(
            false, a0, false, b, (short)0, acc[0][nt], false, false);
        acc[1][nt] = __builtin_amdgcn_wmma_f32_16x16x4_f32(
            false, a1, false, b, (short)0, acc[1][nt], false, false);
      }
    }
  }

  // ---- epilogue: Out = alpha*acc + beta*Cin ----
#pragma unroll
  for (int mt = 0; mt < 2; ++mt) {
#pragma unroll
    for (int nt = 0; nt < 4; ++nt) {
#pragma unroll
      for (int r = 0; r < 8; ++r) {
        const size_t row = bm + wm * 32 + mt * 16 + r + 8 * lh;
        const size_t col = bn + wn * 64 + nt * 16 + l16;
        float v = alpha * acc[mt][nt][r];
        if (beta != 0.0f) v += beta * Cin[row * NDIM + col];
        Out[row * NDIM + col] = v;
      }
    }
  }
}

// out[row] = base + (sub ? sub - acc : acc), acc = M1@v1 + M2@v2 + bias.
__global__ __launch_bounds__(256) void gemv_fused(const float* __restrict__ M1,
                                                  const float* __restrict__ v1,
                                                  const float* __restrict__ M2,
                                                  const float* __restrict__ v2,
                                                  const float* __restrict__ bias,
                                                  const float* __restrict__ sub,
                                                  const float* __restrict__ base,
                                                  float* __restrict__ out) {
  __shared__ float red[8];
  const int row = blockIdx.x;
  const int tid = threadIdx.x;
  float s = 0.0f;
  const float* r1 = M1 + (size_t)row * NDIM;
  for (int j = tid; j < NDIM; j += 256) s += r1[j] * v1[j];
  if (M2) {
    const float* r2 = M2 + (size_t)row * NDIM;
    for (int j = tid; j < NDIM; j += 256) s += r2[j] * v2[j];
  }
  for (int off = 16; off; off >>= 1) s += __shfl_down(s, off, 32);
  if ((tid & 31) == 0) red[tid >> 5] = s;
  __syncthreads();
  if (tid == 0) {
    float t = 0.0f;
    for (int w = 0; w < 8; ++w) t += red[w];
    if (bias) t += bias[row];
    float r = sub ? (sub[row] - t) : t;
    if (base) r += base[row];
    out[row] = r;
  }
}

__global__ __launch_bounds__(256) void row_abs_sum(const float* __restrict__ M,
                                                   float* __restrict__ out) {
  __shared__ float red[8];
  const int row = blockIdx.x;
  const int tid = threadIdx.x;
  float s = 0.0f;
  const float* r1 = M + (size_t)row * NDIM;
  for (int j = tid; j < NDIM; j += 256) s += fabsf(r1[j]);
  for (int off = 16; off; off >>= 1) s += __shfl_down(s, off, 32);
  if ((tid & 31) == 0) red[tid >> 5] = s;
  __syncthreads();
  if (tid == 0) {
    float t = 0.0f;
    for (int w = 0; w < 8; ++w) t += red[w];
    out[row] = t;
  }
}

// sc[0] = 1 / max(rowsum)^2   (Newton-Schulz init scale; Py symmetric)
__global__ __launch_bounds__(256) void max_recip_sq(const float* __restrict__ rs,
                                                    float* __restrict__ sc) {
  __shared__ float red[8];
  const int tid = threadIdx.x;
  float m = 0.0f;
  for (int j = tid; j < NDIM; j += 256) m = fmaxf(m, rs[j]);
  for (int off = 16; off; off >>= 1) m = fmaxf(m, __shfl_down(m, off, 32));
  if ((tid & 31) == 0) red[tid >> 5] = m;
  __syncthreads();
  if (tid == 0) {
    float t = 0.0f;
    for (int w = 0; w < 8; ++w) t = fmaxf(t, red[w]);
    sc[0] = 1.0f / (t * t);
  }
}

__global__ __launch_bounds__(256) void scale_by(const float* __restrict__ in,
                                                const float* __restrict__ sc,
                                                float* __restrict__ out) {
  const float s = sc[0];
  const size_t n = (size_t)NDIM * NDIM;
  for (size_t i = (size_t)blockIdx.x * 256 + threadIdx.x; i < n;
       i += (size_t)gridDim.x * 256)
    out[i] = in[i] * s;
}

__global__ __launch_bounds__(256) void add_mat(const float* __restrict__ a,
                                               const float* __restrict__ b,
                                               float* __restrict__ out) {
  const size_t n = (size_t)NDIM * NDIM;
  for (size_t i = (size_t)blockIdx.x * 256 + threadIdx.x; i < n;
       i += (size_t)gridDim.x * 256)
    out[i] = a[i] + b[i];
}

extern "C" void kernel_launch(void* const* d_in, const int* in_sizes, int n_in,
                              void* d_out, int out_size, void* d_ws, size_t ws_size,
                              hipStream_t stream) {
  (void)in_sizes; (void)n_in; (void)out_size; (void)ws_size;
  const float* x  = (const float*)d_in[0];
  const float* y  = (const float*)d_in[1];
  const float* u  = (const float*)d_in[2];
  const float* P  = (const float*)d_in[3];
  const float* Q  = (const float*)d_in[4];
  const float* R  = (const float*)d_in[5];
  const float* A  = (const float*)d_in[6];
  const float* B  = (const float*)d_in[7];
  const float* C  = (const float*)d_in[8];
  const float* D  = (const float*)d_in[9];
  const float* c1 = (const float*)d_in[10];
  const float* c2 = (const float*)d_in[11];
  float* outp = (float*)d_out;           // [0..N) = x_out, [N..N+N*N) = P_out

  const size_t MS = (size_t)NDIM * NDIM;
  float* B0 = (float*)d_ws;      // Pk
  float* B1 = B0 + MS;           // T1 / Pxy / T2
  float* B2 = B1 + MS;           // Py
  float* B3 = B2 + MS;           // X (ping)
  float* B4 = B3 + MS;           // T / K
  float* B5 = B4 + MS;           // X (pong)
  float* vx = B5 + MS;           // x_pred
  float* vy = vx + NDIM;         // y residual
  float* rs = vy + NDIM;         // row |sums|
  float* sc = rs + NDIM;         // scalar 1/||Py||inf^2

  const dim3 gg(NDIM / BT, NDIM / BT);
  const dim3 gb(256);

  // x_pred = A@x + B@u + c1
  gemv_fused<<<NDIM, gb, 0, stream>>>(A, x, B, u, c1, nullptr, nullptr, vx);
  // Pk = P + Q  (symmetric)
  add_mat<<<2048, gb, 0, stream>>>(P, Q, B0);
  // T1 = C @ Pk = C @ Pk^T
  gemm_wmma_nt<<<gg, gb, 0, stream>>>(C, B0, nullptr, B1, 1.0f, 0.0f);
  // Py = T1 @ C^T + R
  gemm_wmma_nt<<<gg, gb, 0, stream>>>(B1, C, R, B2, 1.0f, 1.0f);
  // Pxy = Pk @ C^T
  gemm_wmma_nt<<<gg, gb, 0, stream>>>(B0, C, nullptr, B1, 1.0f, 0.0f);
  // vy = y - (C@x_pred + D@u + c2)
  gemv_fused<<<NDIM, gb, 0, stream>>>(C, vx, D, u, c2, y, nullptr, vy);

  // X0 = Py / ||Py||inf^2  (Py symmetric SPD)
  row_abs_sum<<<NDIM, gb, 0, stream>>>(B2, rs);
  max_recip_sq<<<1, gb, 0, stream>>>(rs, sc);
  scale_by<<<2048, gb, 0, stream>>>(B2, sc, B3);

  // Newton-Schulz: X <- 2X - X@Py@X ; X and T = Py@X stay symmetric
  float* Xa = B3;
  float* Xb = B5;
  for (int it = 0; it < 24; ++it) {
    gemm_wmma_nt<<<gg, gb, 0, stream>>>(B2, Xa, nullptr, B4, 1.0f, 0.0f);  // T  = Py@X^T
    gemm_wmma_nt<<<gg, gb, 0, stream>>>(Xa, B4, Xa, Xb, -1.0f, 2.0f);      // X' = 2X - X@T^T
    float* t = Xa; Xa = Xb; Xb = t;
  }

  // K = Pxy @ Py^-1 = Pxy @ X^T
  gemm_wmma_nt<<<gg, gb, 0, stream>>>(B1, Xa, nullptr, B4, 1.0f, 0.0f);
  // T2 = K @ Py = K @ Py^T
  gemm_wmma_nt<<<gg, gb, 0, stream>>>(B4, B2, nullptr, B1, 1.0f, 0.0f);
  // P_out = Pk - T2 @ K^T
  gemm_wmma_nt<<<gg, gb, 0, stream>>>(B1, B4, B0, outp + NDIM, -1.0f, 1.0f);
  // x_out = x_pred + K @ vy
  gemv_fused<<<NDIM, gb, 0, stream>>>(B4, vy, nullptr, nullptr, nullptr, nullptr,
                                      vx, outp);
}